// ConvLSTM_82119774699861
// MI455X (gfx1250) — compile-verified
//
#include <hip/hip_runtime.h>
#include <hip/hip_bf16.h>
#include <stdint.h>

// ConvLSTM forward on MI455X (gfx1250).
// Live computation after DCE of it/ft/Ct:
//   out[n,h,w] = sigmoid( conv3x3(x[n],W8)+b8 + conv3x3(H0,W9)+b9
//                         + conv3x3(C0,W10)+b10 + bgate[3,h,w] )
// Bandwidth-bound (~268 MB @ 23.3 TB/s): use CDNA5 async global->LDS staging
// for the x tiles, vectorized LDS reads, and non-temporal output stores.

#define GAS __attribute__((address_space(1)))
#define LAS __attribute__((address_space(3)))

typedef __attribute__((ext_vector_type(4))) float f32x4;
typedef __attribute__((ext_vector_type(2))) float f32x2;

#define HW      1024
#define TILE_W  128
#define TILE_H  16
#define LDS_W   132   // 130 used + 2 pad; multiple of 4 -> 16B-aligned b128 LDS reads
#define LDS_H   18

__device__ __forceinline__ void async_ld_f32(const float* gp, float* lp) {
#if __has_builtin(__builtin_amdgcn_global_load_async_to_lds_b32)
    // gfx1250 GLOBAL_LOAD_ASYNC_TO_LDS_B32 (ASYNCcnt-tracked DMA to LDS).
    // Builtin signature wants (as1 int*, as3 int*, imm offset, imm cpol).
    float* gnc = const_cast<float*>(gp);
    __builtin_amdgcn_global_load_async_to_lds_b32((GAS int*)gnc, (LAS int*)lp,
                                                  0, 0);
#else
    *lp = *gp;  // fallback: global_load + ds_store
#endif
}

__device__ __forceinline__ void wait_async_zero() {
#if __has_builtin(__builtin_amdgcn_s_wait_asynccnt)
    __builtin_amdgcn_s_wait_asynccnt(0);
#else
    asm volatile("s_wait_asynccnt 0x0" ::: "memory");
#endif
}

__device__ __forceinline__ float sigmoid_fast(float v) {
    return 1.0f / (1.0f + __expf(-v));
}

// Pass A: S[h,w] = conv(H0,W9)+conv(C0,W10) + b9 + b10 + bgate[3,h,w]
__global__ __launch_bounds__(256) void convlstm_S_kernel(
    const float* __restrict__ H0, const float* __restrict__ C0,
    const float* __restrict__ Wconv, const float* __restrict__ bconv,
    const float* __restrict__ bgate, float* __restrict__ S)
{
    const int idx = blockIdx.x * 256 + threadIdx.x;   // one output each
    const int h = idx >> 10, w = idx & (HW - 1);
    const float* w9  = Wconv + 9 * 9;    // Who
    const float* w10 = Wconv + 10 * 9;   // Wco
    float acc = bconv[9] + bconv[10] + bgate[3 * HW * HW + idx];
#pragma unroll
    for (int dy = 0; dy < 3; ++dy) {
        const int hh = h + dy - 1;
        if ((unsigned)hh < (unsigned)HW) {
#pragma unroll
            for (int dx = 0; dx < 3; ++dx) {
                const int ww = w + dx - 1;
                if ((unsigned)ww < (unsigned)HW) {
                    const int o = hh * HW + ww;
                    acc = fmaf(H0[o], w9[dy * 3 + dx], acc);
                    acc = fmaf(C0[o], w10[dy * 3 + dx], acc);
                }
            }
        }
    }
    S[idx] = acc;
}

// Pass B: out[n] = sigmoid(conv3x3(x[n],W8) + b8 + S), tiled through LDS.
__global__ __launch_bounds__(256) void convlstm_gate_kernel(
    const float* __restrict__ x, const float* __restrict__ Wconv,
    const float* __restrict__ bconv, const float* __restrict__ S,
    float* __restrict__ out)
{
    __shared__ float tile[LDS_H * LDS_W];
    const int tid = threadIdx.x;
    const int tw0 = blockIdx.x * TILE_W;
    const int th0 = blockIdx.y * TILE_H;
    const int n   = blockIdx.z;
    const float* xp = x + (size_t)n * (HW * HW);

    // Stage haloed 130x18 x-tile into LDS: async DMA for in-bounds elements,
    // ds_store zeros for the 'same' padding ring / stride pad.
    for (int i = tid; i < LDS_H * LDS_W; i += 256) {
        const int r = i / LDS_W, c = i % LDS_W;
        const int gh = th0 - 1 + r;
        const int gw = tw0 - 1 + c;
        const bool valid = (c < 130) & ((unsigned)gh < (unsigned)HW) &
                           ((unsigned)gw < (unsigned)HW);
        if (valid) {
            async_ld_f32(xp + gh * HW + gw, &tile[i]);
        } else {
            tile[i] = 0.0f;
        }
    }
    wait_async_zero();     // ASYNCcnt == 0: this wave's DMAs landed in LDS
    __syncthreads();       // all 8 waves' tiles visible

    const float* w8 = Wconv + 8 * 9;     // Wxo (uniform -> scalar loads)
    const float k0 = w8[0], k1 = w8[1], k2 = w8[2];
    const float k3 = w8[3], k4 = w8[4], k5 = w8[5];
    const float k6 = w8[6], k7 = w8[7], k8 = w8[8];
    const float b8 = bconv[8];

    // Thread -> 4 adjacent columns x 2 rows (8 outputs).
    const int tx4 = (tid & 31) * 4;      // 0..124
    const int ty  = tid >> 5;            // 0..7
#pragma unroll
    for (int rr = 0; rr < 2; ++rr) {
        const int row = ty + rr * 8;     // 0..15
        float v[3][6];
#pragma unroll
        for (int dy = 0; dy < 3; ++dy) { // 3 kernel rows: b128 + b64 LDS reads
            const float* base = &tile[(row + dy) * LDS_W + tx4];
            const f32x4 a = *(const f32x4*)base;
            const f32x2 b = *(const f32x2*)(base + 4);
            v[dy][0] = a.x; v[dy][1] = a.y; v[dy][2] = a.z;
            v[dy][3] = a.w; v[dy][4] = b.x; v[dy][5] = b.y;
        }
        const int gh = th0 + row;
        const size_t so = (size_t)gh * HW + tw0 + tx4;
        const f32x4 s4 = *(const f32x4*)(S + so);
        f32x4 o;
#pragma unroll
        for (int j = 0; j < 4; ++j) {
            float acc = b8 + s4[j];
            acc = fmaf(v[0][j],     k0, acc);
            acc = fmaf(v[0][j + 1], k1, acc);
            acc = fmaf(v[0][j + 2], k2, acc);
            acc = fmaf(v[1][j],     k3, acc);
            acc = fmaf(v[1][j + 1], k4, acc);
            acc = fmaf(v[1][j + 2], k5, acc);
            acc = fmaf(v[2][j],     k6, acc);
            acc = fmaf(v[2][j + 1], k7, acc);
            acc = fmaf(v[2][j + 2], k8, acc);
            o[j] = sigmoid_fast(acc);
        }
        // Streamed 128 MB output: non-temporal, keep L2 for x/S reuse.
        __builtin_nontemporal_store(o, (f32x4*)(out + (size_t)n * HW * HW + so));
    }
}

extern "C" void kernel_launch(void* const* d_in, const int* in_sizes, int n_in,
                              void* d_out, int out_size, void* d_ws, size_t ws_size,
                              hipStream_t stream) {
    (void)n_in; (void)out_size; (void)ws_size;
    const float* x     = (const float*)d_in[0];
    const float* H0    = (const float*)d_in[1];
    const float* C0    = (const float*)d_in[2];
    const float* Wconv = (const float*)d_in[3];
    const float* bconv = (const float*)d_in[4];
    const float* bgate = (const float*)d_in[5];
    float* out = (float*)d_out;
    float* S   = (float*)d_ws;                      // 1024*1024 f32 = 4 MB scratch

    const int B = in_sizes[0] / (HW * HW);          // 32

    convlstm_S_kernel<<<(HW * HW) / 256, 256, 0, stream>>>(H0, C0, Wconv, bconv,
                                                           bgate, S);
    dim3 grid(HW / TILE_W, HW / TILE_H, B);         // 8 x 64 x 32 blocks
    convlstm_gate_kernel<<<grid, 256, 0, stream>>>(x, Wconv, bconv, S, out);
}